// DenseSAKELayer_42537356100223
// MI455X (gfx1250) — compile-verified
//
#include <hip/hip_runtime.h>
#include <math.h>

typedef __attribute__((ext_vector_type(2))) float v2f;
typedef __attribute__((ext_vector_type(8))) float v8f;

#define NN   256
#define FF   64
#define HH   64
#define NHEAD 4
#define KRBF 50
#define CC   256

__device__ __forceinline__ float siluf(float x) { return x / (1.f + __expf(-x)); }

__device__ __forceinline__ void softmax_j(float* arr, float* red, int tid) {
  // arr: [NN][NHEAD], softmax over j (first index), per head
  __syncthreads();
  if (tid < NHEAD) {
    float mx = -3.0e38f;
    for (int j = 0; j < NN; ++j) mx = fmaxf(mx, arr[j * NHEAD + tid]);
    red[tid] = mx;
  }
  __syncthreads();
  for (int hd = 0; hd < NHEAD; ++hd)
    arr[tid * NHEAD + hd] = __expf(arr[tid * NHEAD + hd] - red[hd]);
  __syncthreads();
  if (tid < NHEAD) {
    float s = 0.f;
    for (int j = 0; j < NN; ++j) s += arr[j * NHEAD + tid];
    red[4 + tid] = s;
  }
  __syncthreads();
  for (int hd = 0; hd < NHEAD; ++hd)
    arr[tid * NHEAD + hd] *= (1.f / red[4 + hd]);
  __syncthreads();
}

#define WMMA_F32(A, B, C) \
  __builtin_amdgcn_wmma_f32_16x16x4_f32(false, (A), false, (B), (short)0, (C), false, false)

__global__ __launch_bounds__(256) void sake_edge_kernel(
    const float* __restrict__ h, const float* __restrict__ x,
    const float* __restrict__ means, const float* __restrict__ betas,
    const float* __restrict__ W_in, const float* __restrict__ b_in,
    const float* __restrict__ W_o1, const float* __restrict__ b_o1,
    const float* __restrict__ W_o2, const float* __restrict__ b_o2,
    const float* __restrict__ Ws, const float* __restrict__ bs,
    const float* __restrict__ log_gamma, const float* __restrict__ Wx,
    const float* __restrict__ Wv_mix,
    float* __restrict__ ws_cn, float* __restrict__ ws_he, float* __restrict__ ws_dv) {
  __shared__ float bufB[NN * 64];  // hk -> rbf*hk -> h_e (edge features)
  __shared__ float bufC[NN * 64];  // h1 -> silu(h1)
  __shared__ float norm_s[NN];
  __shared__ float xd_s[NN * 3];
  __shared__ float attE[NN * NHEAD], attS[NN * NHEAD], attC[NN * NHEAD];
  __shared__ float constk[64], cst1[64];
  __shared__ float comb_sum[CC * 3];
  __shared__ float s_lds[NN];
  __shared__ float red[8];

  const int tid = threadIdx.x;
  const int lane = tid & 31, wave = tid >> 5;
  const int lo = lane & 15, hi = lane >> 4;
  const int row = blockIdx.x;  // b*256 + i
  const int b = row >> 8, i = row & 255;
  const float* hg = h + (size_t)b * NN * FF;
  const float* xg = x + (size_t)b * NN * 3;

  // ---- phase 1: geometry per j ----
  {
    int j = tid;
    float dx0 = xg[j * 3 + 0] - xg[i * 3 + 0];
    float dx1 = xg[j * 3 + 1] - xg[i * 3 + 1];
    float dx2 = xg[j * 3 + 2] - xg[i * 3 + 2];
    float d2 = dx0 * dx0 + dx1 * dx1 + dx2 * dx2;
    float nr = sqrtf(d2 + 1e-5f);
    norm_s[j] = nr;
    float inv = 1.f / (nr + 1e-5f);
    xd_s[j * 3 + 0] = dx0 * inv;
    xd_s[j * 3 + 1] = dx1 * inv;
    xd_s[j * 3 + 2] = dx2 * inv;
  }
  // ---- phase 2: row-constant vectors (h_i contributions) ----
  if (tid < KRBF) {
    float s = b_in[tid];
    for (int f = 0; f < FF; ++f) s += hg[i * FF + f] * W_in[(FF + f) * KRBF + tid];
    constk[tid] = s;
  }
  if (tid >= 64 && tid < 128) {
    int n = tid - 64;
    float s = b_o1[n];
    for (int f = 0; f < FF; ++f) s += hg[i * FF + f] * W_o1[(64 + f) * HH + n];
    cst1[n] = s;
  }
  __syncthreads();

  // ---- GEMM1: hk[j,k] = h_j @ W_in[0:64,:]  (M=256,K=64,N=50->64) ----
  for (int q = 0; q < 8; ++q) {
    int t = wave * 8 + q;
    int j0 = (t >> 2) * 16, n0 = (t & 3) * 16;
    int n = n0 + lo, jr = j0 + lo;
    v8f acc = {0.f, 0.f, 0.f, 0.f, 0.f, 0.f, 0.f, 0.f};
    for (int k0 = 0; k0 < 64; k0 += 4) {
      int ka = k0 + 2 * hi;
      v2f a, bb;
      a[0] = hg[jr * FF + ka];
      a[1] = hg[jr * FF + ka + 1];
      bb[0] = (n < KRBF) ? W_in[ka * KRBF + n] : 0.f;
      bb[1] = (n < KRBF) ? W_in[(ka + 1) * KRBF + n] : 0.f;
      acc = WMMA_F32(a, bb, acc);
    }
    for (int vv = 0; vv < 8; ++vv) bufB[(j0 + vv + 8 * hi) * 64 + n] = acc[vv];
  }
  __syncthreads();

  // ---- rbf * (hk + const_k) ----
  {
    int j = tid;
    float en = __expf(-norm_s[j]);
    for (int k = 0; k < KRBF; ++k) {
      float hk = bufB[j * 64 + k] + constk[k];
      float d = en - means[k];
      bufB[j * 64 + k] = __expf(-betas[k] * d * d) * hk;
    }
    for (int k = KRBF; k < 64; ++k) bufB[j * 64 + k] = 0.f;
  }
  __syncthreads();

  // ---- GEMM2: h1 = h_j @ Wo1[0:64] + rbfhk @ Wo1[128:178]  (K=128 eff) ----
  for (int q = 0; q < 8; ++q) {
    int t = wave * 8 + q;
    int j0 = (t >> 2) * 16, n0 = (t & 3) * 16;
    int n = n0 + lo, jr = j0 + lo;
    v8f acc = {0.f, 0.f, 0.f, 0.f, 0.f, 0.f, 0.f, 0.f};
    for (int k0 = 0; k0 < 128; k0 += 4) {
      int ka = k0 + 2 * hi;
      v2f a, bb;
      if (k0 < 64) {
        a[0] = hg[jr * FF + ka];
        a[1] = hg[jr * FF + ka + 1];
        bb[0] = W_o1[ka * HH + n];
        bb[1] = W_o1[(ka + 1) * HH + n];
      } else {
        int kk = ka - 64;
        a[0] = bufB[jr * 64 + kk];
        a[1] = bufB[jr * 64 + kk + 1];
        bb[0] = (kk < KRBF) ? W_o1[(128 + kk) * HH + n] : 0.f;
        bb[1] = (kk + 1 < KRBF) ? W_o1[(128 + kk + 1) * HH + n] : 0.f;
      }
      acc = WMMA_F32(a, bb, acc);
    }
    for (int vv = 0; vv < 8; ++vv) bufC[(j0 + vv + 8 * hi) * 64 + n] = acc[vv];
  }
  __syncthreads();
  // add norm * Wo1[178] + cst1, then silu (in place)
  {
    int j = tid;
    float nr = norm_s[j];
    for (int n = 0; n < HH; ++n) {
      float t = bufC[j * 64 + n] + nr * W_o1[178 * HH + n] + cst1[n];
      bufC[j * 64 + n] = siluf(t);
    }
  }
  __syncthreads();

  // ---- GEMM3: h_e = silu(h1) @ W_o2 + b_o2 ----
  for (int q = 0; q < 8; ++q) {
    int t = wave * 8 + q;
    int j0 = (t >> 2) * 16, n0 = (t & 3) * 16;
    int n = n0 + lo, jr = j0 + lo;
    v8f acc = {0.f, 0.f, 0.f, 0.f, 0.f, 0.f, 0.f, 0.f};
    for (int k0 = 0; k0 < 64; k0 += 4) {
      int ka = k0 + 2 * hi;
      v2f a, bb;
      a[0] = bufC[jr * 64 + ka];
      a[1] = bufC[jr * 64 + ka + 1];
      bb[0] = W_o2[ka * HH + n];
      bb[1] = W_o2[(ka + 1) * HH + n];
      acc = WMMA_F32(a, bb, acc);
    }
    float bo2 = b_o2[n];
    for (int vv = 0; vv < 8; ++vv) bufB[(j0 + vv + 8 * hi) * 64 + n] = acc[vv] + bo2;
  }
  __syncthreads();

  // ---- attentions: logits ----
  {
    int j = tid;
    float eyepen = (j == i) ? 1e5f : 0.f;
    for (int hd = 0; hd < NHEAD; ++hd) {
      float s = bs[hd];
      for (int m = 0; m < HH; ++m) s += bufB[j * 64 + m] * Ws[m * NHEAD + hd];
      float c = (s > 0.f) ? s : 2.f * (__expf(s * 0.5f) - 1.f);  // celu alpha=2
      attS[j * NHEAD + hd] = c - eyepen;
      attE[j * NHEAD + hd] = -(norm_s[j] + eyepen) * __expf(log_gamma[hd]);
    }
  }
  softmax_j(attE, red, tid);
  softmax_j(attS, red, tid);
  for (int hd = 0; hd < NHEAD; ++hd)
    attC[tid * NHEAD + hd] = attE[tid * NHEAD + hd] * attS[tid * NHEAD + hd];
  softmax_j(attC, red, tid);

  // ---- h_e aggregate over j; zero accumulators ----
  {
    int c = tid;
    float s = 0.f;
    for (int j = 0; j < NN; ++j)
      s += bufB[j * 64 + (c >> 2)] * attC[j * NHEAD + (c & 3)];
    ws_he[(size_t)row * CC + c] = s;
    s_lds[tid] = 0.f;
    comb_sum[tid * 3 + 0] = 0.f;
    comb_sum[tid * 3 + 1] = 0.f;
    comb_sum[tid * 3 + 2] = 0.f;
  }
  __syncthreads();

  // ---- GEMM4 (fused): coeff = tanh(h_e_att @ Wx); reduce over j on the fly ----
  for (int nt = wave * 2; nt < wave * 2 + 2; ++nt) {
    int n0 = nt * 16, n = n0 + lo;
    float wvc = Wv_mix[n];
    float cs0 = 0.f, cs1 = 0.f, cs2 = 0.f;
    for (int jt = 0; jt < 16; ++jt) {
      int j0 = jt * 16, jr = j0 + lo;
      v8f acc = {0.f, 0.f, 0.f, 0.f, 0.f, 0.f, 0.f, 0.f};
      for (int k0 = 0; k0 < CC; k0 += 4) {
        int ka = k0 + 2 * hi;
        v2f a, bb;
        // h_e_att[j,c] = h_e[j, c>>2] * comb_att[j, c&3]
        a[0] = bufB[jr * 64 + (ka >> 2)] * attC[jr * NHEAD + (ka & 3)];
        a[1] = bufB[jr * 64 + ((ka + 1) >> 2)] * attC[jr * NHEAD + ((ka + 1) & 3)];
        bb[0] = Wx[ka * CC + n];
        bb[1] = Wx[(ka + 1) * CC + n];
        acc = WMMA_F32(a, bb, acc);
      }
      for (int vv = 0; vv < 8; ++vv) {
        float cf = tanhf(acc[vv]);
        int j = j0 + vv + 8 * hi;
        cs0 += xd_s[j * 3 + 0] * cf;
        cs1 += xd_s[j * 3 + 1] * cf;
        cs2 += xd_s[j * 3 + 2] * cf;
        float sv = cf * wvc;  // sum over c' of coeff * Wv_mix
        sv += __shfl_xor(sv, 1, 32);
        sv += __shfl_xor(sv, 2, 32);
        sv += __shfl_xor(sv, 4, 32);
        sv += __shfl_xor(sv, 8, 32);
        if (lo == 0) atomicAdd(&s_lds[j], sv);
      }
    }
    atomicAdd(&comb_sum[n * 3 + 0], cs0);
    atomicAdd(&comb_sum[n * 3 + 1], cs1);
    atomicAdd(&comb_sum[n * 3 + 2], cs2);
  }
  __syncthreads();

  // ---- finalize per-row outputs ----
  {
    int c = tid;
    float a0 = comb_sum[c * 3 + 0] * (1.f / NN);
    float a1 = comb_sum[c * 3 + 1] * (1.f / NN);
    float a2 = comb_sum[c * 3 + 2] * (1.f / NN);
    ws_cn[(size_t)row * CC + c] = a0 * a0 + a1 * a1 + a2 * a2;
  }
  if (tid < 3) {
    float d = 0.f;
    for (int j = 0; j < NN; ++j) d += xd_s[j * 3 + tid] * s_lds[j];
    ws_dv[row * 3 + tid] = d * (1.f / NN);
  }
}

__global__ __launch_bounds__(64) void sake_node_kernel(
    const float* __restrict__ h, const float* __restrict__ x, const float* __restrict__ v,
    const float* __restrict__ Wp1, const float* __restrict__ bp1,
    const float* __restrict__ Wp2, const float* __restrict__ bp2,
    const float* __restrict__ Wn1, const float* __restrict__ bn1,
    const float* __restrict__ Wn2, const float* __restrict__ bn2,
    const float* __restrict__ Wvel1, const float* __restrict__ bvel1,
    const float* __restrict__ Wvel2,
    const float* __restrict__ ws_cn, const float* __restrict__ ws_he,
    const float* __restrict__ ws_dv, float* __restrict__ out) {
  __shared__ float cn[CC], inb[384], g1[64], u1[64], hnew[64], w1[64];
  const int tid = threadIdx.x;
  const int row = blockIdx.x;  // b*256 + i
  for (int q = 0; q < 4; ++q) {
    cn[tid * 4 + q] = ws_cn[(size_t)row * CC + tid * 4 + q];
    inb[64 + tid * 4 + q] = ws_he[(size_t)row * CC + tid * 4 + q];
  }
  inb[tid] = h[row * 64 + tid];
  __syncthreads();
  {  // post_norm_mlp layer 1: 256 -> 64
    float s = bp1[tid];
    for (int c = 0; c < CC; ++c) s += cn[c] * Wp1[c * 64 + tid];
    g1[tid] = siluf(s);
  }
  __syncthreads();
  {  // post_norm_mlp layer 2: 64 -> 64, result into inb[320:384]
    float s = bp2[tid];
    for (int m = 0; m < 64; ++m) s += g1[m] * Wp2[m * 64 + tid];
    inb[320 + tid] = siluf(s);
  }
  __syncthreads();
  {  // node_mlp layer 1: 384 -> 64
    float s = bn1[tid];
    for (int r = 0; r < 384; ++r) s += inb[r] * Wn1[r * 64 + tid];
    u1[tid] = siluf(s);
  }
  __syncthreads();
  {  // node_mlp layer 2: 64 -> 64, residual
    float s = bn2[tid];
    for (int m = 0; m < 64; ++m) s += u1[m] * Wn2[m * 64 + tid];
    float hn = h[row * 64 + tid] + siluf(s);
    hnew[tid] = hn;
    out[row * 64 + tid] = hn;
  }
  __syncthreads();
  {  // velocity mlp layer 1: 64 -> 64
    float s = bvel1[tid];
    for (int f = 0; f < 64; ++f) s += hnew[f] * Wvel1[f * 64 + tid];
    w1[tid] = siluf(s);
  }
  __syncthreads();
  if (tid == 0) {
    float sc = 0.f;
    for (int m = 0; m < 64; ++m) sc += w1[m] * Wvel2[m];
    float vs = 2.f / (1.f + __expf(-sc));
    const int XOFF = 2 * 256 * 64;        // 32768
    const int VOFF = XOFF + 2 * 256 * 3;  // 34304
    for (int t = 0; t < 3; ++t) {
      float dv = ws_dv[row * 3 + t];
      float vn = dv + vs * v[row * 3 + t];
      out[XOFF + row * 3 + t] = x[row * 3 + t] + vn;
      out[VOFF + row * 3 + t] = vn;
    }
  }
}

extern "C" void kernel_launch(void* const* d_in, const int* in_sizes, int n_in,
                              void* d_out, int out_size, void* d_ws, size_t ws_size,
                              hipStream_t stream) {
  const float* h        = (const float*)d_in[0];
  const float* x        = (const float*)d_in[1];
  const float* v        = (const float*)d_in[2];
  const float* means    = (const float*)d_in[3];
  const float* betas    = (const float*)d_in[4];
  const float* W_in     = (const float*)d_in[5];
  const float* b_in     = (const float*)d_in[6];
  const float* W_o1     = (const float*)d_in[7];
  const float* b_o1     = (const float*)d_in[8];
  const float* W_o2     = (const float*)d_in[9];
  const float* b_o2     = (const float*)d_in[10];
  const float* Ws       = (const float*)d_in[11];
  const float* bs       = (const float*)d_in[12];
  const float* log_gamma= (const float*)d_in[13];
  const float* Wx       = (const float*)d_in[14];
  const float* Wp1      = (const float*)d_in[15];
  const float* bp1      = (const float*)d_in[16];
  const float* Wp2      = (const float*)d_in[17];
  const float* bp2      = (const float*)d_in[18];
  const float* Wn1      = (const float*)d_in[19];
  const float* bn1      = (const float*)d_in[20];
  const float* Wn2      = (const float*)d_in[21];
  const float* bn2      = (const float*)d_in[22];
  const float* Wv_mix   = (const float*)d_in[23];
  const float* Wvel1    = (const float*)d_in[24];
  const float* bvel1    = (const float*)d_in[25];
  const float* Wvel2    = (const float*)d_in[26];
  float* out = (float*)d_out;
  float* ws = (float*)d_ws;
  float* ws_cn = ws;                  // [512][256]
  float* ws_he = ws + 512 * 256;      // [512][256]
  float* ws_dv = ws + 2 * 512 * 256;  // [512][3]

  sake_edge_kernel<<<dim3(512), dim3(256), 0, stream>>>(
      h, x, means, betas, W_in, b_in, W_o1, b_o1, W_o2, b_o2, Ws, bs,
      log_gamma, Wx, Wv_mix, ws_cn, ws_he, ws_dv);
  sake_node_kernel<<<dim3(512), dim3(64), 0, stream>>>(
      h, x, v, Wp1, bp1, Wp2, bp2, Wn1, bn1, Wn2, bn2, Wvel1, bvel1, Wvel2,
      ws_cn, ws_he, ws_dv, out);
}